// CochlearEncoder_39032662786217
// MI455X (gfx1250) — compile-verified
//
#include <hip/hip_runtime.h>
#include <hip/hip_bf16.h>
#include <math.h>

#define NFFT   512
#define NBINS  257
#define NFILT  128
#define OUTN   262144
#define NTS    128
#define TWO_PI_OVER_N 0.01227184630308513f   // 2*pi/512

typedef __attribute__((ext_vector_type(2))) float v2f;
typedef __attribute__((ext_vector_type(8))) float v8f;

// ws layout (float units):
//   [0,128)             adapted vector
//   [128],[129]         encoded global min / max (uint32)
//   [256, 256+OUTN)     activity vector

__device__ __forceinline__ unsigned int enc_f32(float f) {
    unsigned int u = __float_as_uint(f);
    return (u & 0x80000000u) ? ~u : (u | 0x80000000u);
}
__device__ __forceinline__ float dec_f32(unsigned int u) {
    return (u & 0x80000000u) ? __uint_as_float(u ^ 0x80000000u)
                             : __uint_as_float(~u);
}

// ---------------------------------------------------------------------------
// Kernel 1 (single block): DFT magnitude -> mel filterbank -> hair-cell
// compression -> adapted[128] in ws; also init min/max atomic slots.
// ---------------------------------------------------------------------------
__global__ __launch_bounds__(256) void k1_front(const float* __restrict__ audio,
                                                const float* __restrict__ fb,
                                                const float* __restrict__ adapt_state,
                                                float* __restrict__ ws_f)
{
    __shared__ float s_audio[NFFT];
    __shared__ float s_mag[NBINS];
    const int tid = threadIdx.x;

    for (int i = tid; i < NFFT; i += 256) s_audio[i] = audio[i];
    __syncthreads();

    // Naive DFT, phase reduced mod 2*pi exactly via (k*n) & 511
    for (int k = tid; k < NBINS; k += 256) {
        float re = 0.f, im = 0.f;
        for (int n = 0; n < NFFT; ++n) {
            int   t   = (k * n) & (NFFT - 1);
            float ang = (float)t * TWO_PI_OVER_N;
            float s, c;
            __sincosf(ang, &s, &c);
            float a = s_audio[n];
            re += a * c;
            im -= a * s;
        }
        s_mag[k] = sqrtf(re * re + im * im);
    }
    __syncthreads();

    for (int f = tid; f < NFILT; f += 256) {
        float acc = 0.f;
        const float* row = fb + f * NBINS;
        for (int j = 0; j < NBINS; ++j) acc += row[j] * s_mag[j];
        float hair    = powf(acc + 1e-6f, 0.3f);
        float adapted = hair - adapt_state[f] * 0.5f;
        ws_f[f] = adapted > 0.f ? adapted : 0.f;
    }

    if (tid == 0) {
        unsigned int* mm = (unsigned int*)(ws_f + 128);
        mm[0] = 0xFFFFFFFFu;  // encoded min identity
        mm[1] = 0u;           // encoded max identity
    }
}

// ---------------------------------------------------------------------------
// Kernel 2: activity = W @ adapted + b via V_WMMA_F32_16X16X4_F32.
// One wave -> 16 output rows; B columns are 16 broadcast copies of adapted.
// Fused block-level min/max reduction -> global atomics on encoded floats.
// ---------------------------------------------------------------------------
__global__ __launch_bounds__(256) void k2_wmma_matvec(const float* __restrict__ W,
                                                      const float* __restrict__ bias,
                                                      float* __restrict__ ws_f)
{
    __shared__ float s_ad[NFILT];
    __shared__ float s_min[256];
    __shared__ float s_max[256];

    const int tid = threadIdx.x;
    if (tid < NFILT) s_ad[tid] = ws_f[tid];
    __syncthreads();

    const int lane    = tid & 31;        // wave32
    const int wave    = tid >> 5;        // 8 waves/block
    const int half    = lane >> 4;       // 0: K pair {0,1}, 1: K pair {2,3}
    const int m       = lane & 15;       // row within 16-row tile
    const int koff    = half * 2;
    const int rowbase = blockIdx.x * 128 + wave * 16;

    v8f acc = {0.f, 0.f, 0.f, 0.f, 0.f, 0.f, 0.f, 0.f};

    const float* wrow = W + (size_t)(rowbase + m) * NFILT + koff;
    #pragma unroll 4
    for (int kk = 0; kk < NFILT; kk += 4) {
        v2f a = *reinterpret_cast<const v2f*>(wrow + kk);   // global_load_b64
        v2f b;
        b.x = s_ad[kk + koff];
        b.y = s_ad[kk + koff + 1];
        // D = A(16x4) * B(4x16) + C ; all B columns identical -> GEMV
        acc = __builtin_amdgcn_wmma_f32_16x16x4_f32(
                  false, a, false, b, (short)0, acc, false, false);
    }

    // C/D layout: lanes 0-15 hold rows rowbase+0..7 in acc[0..7],
    //             lanes 16-31 hold rows rowbase+8..15.
    const int r0 = rowbase + half * 8;
    float act[8];
    float lmin = 3.402823e38f, lmax = -3.402823e38f;
    #pragma unroll
    for (int j = 0; j < 8; ++j) {
        act[j] = acc[j] + bias[r0 + j];
        lmin = fminf(lmin, act[j]);
        lmax = fmaxf(lmax, act[j]);
    }
    if (m == 0) {                         // lanes 0 and 16 write the 16 rows
        float* out = ws_f + 256 + r0;
        #pragma unroll
        for (int j = 0; j < 8; ++j) out[j] = act[j];
    }

    s_min[tid] = lmin;
    s_max[tid] = lmax;
    __syncthreads();
    for (int s = 128; s > 0; s >>= 1) {
        if (tid < s) {
            s_min[tid] = fminf(s_min[tid], s_min[tid + s]);
            s_max[tid] = fmaxf(s_max[tid], s_max[tid + s]);
        }
        __syncthreads();
    }
    if (tid == 0) {
        unsigned int* mm = (unsigned int*)(ws_f + 128);
        atomicMin(&mm[0], enc_f32(s_min[0]));
        atomicMax(&mm[1], enc_f32(s_max[0]));
    }
}

// ---------------------------------------------------------------------------
// Kernel 3: latency coding -> spikes[t, n] as float {0,1}.
// Each thread owns 4 neurons; 128 coalesced float4-store rounds (134 MB).
// ---------------------------------------------------------------------------
__global__ __launch_bounds__(256) void k3_spikes(const float* __restrict__ ws_f,
                                                 float* __restrict__ out)
{
    const unsigned int* mm = (const unsigned int*)(ws_f + 128);
    const float amin = dec_f32(mm[0]);
    const float amax = dec_f32(mm[1]);
    const float inv  = 1.0f / (amax - amin + 1e-6f);

    const int i = blockIdx.x * blockDim.x + threadIdx.x;   // 0 .. OUTN/4-1
    const float4* actv = reinterpret_cast<const float4*>(ws_f + 256);
    float4 a = actv[i];

    float nx = (a.x - amin) * inv;
    float ny = (a.y - amin) * inv;
    float nz = (a.z - amin) * inv;
    float nw = (a.w - amin) * inv;

    int lx = (int)((1.0f - nx) * (float)(NTS - 1));
    int ly = (int)((1.0f - ny) * (float)(NTS - 1));
    int lz = (int)((1.0f - nz) * (float)(NTS - 1));
    int lw = (int)((1.0f - nw) * (float)(NTS - 1));

    bool fx = nx > 0.05f, fy = ny > 0.05f, fz = nz > 0.05f, fw = nw > 0.05f;

    float4* out4 = reinterpret_cast<float4*>(out);
    const int stride = OUTN / 4;
    #pragma unroll 4
    for (int t = 0; t < NTS; ++t) {
        float4 v;
        v.x = (fx && lx == t) ? 1.0f : 0.0f;
        v.y = (fy && ly == t) ? 1.0f : 0.0f;
        v.z = (fz && lz == t) ? 1.0f : 0.0f;
        v.w = (fw && lw == t) ? 1.0f : 0.0f;
        out4[(size_t)t * stride + i] = v;                  // global_store_b128
    }
}

// ---------------------------------------------------------------------------
extern "C" void kernel_launch(void* const* d_in, const int* in_sizes, int n_in,
                              void* d_out, int out_size, void* d_ws, size_t ws_size,
                              hipStream_t stream) {
    const float* audio = (const float*)d_in[0];   // [16000]
    const float* fb    = (const float*)d_in[1];   // [128*257]
    const float* W     = (const float*)d_in[2];   // [262144*128]
    const float* bias  = (const float*)d_in[3];   // [262144]
    const float* adapt = (const float*)d_in[4];   // [128]
    float* ws_f = (float*)d_ws;
    float* out  = (float*)d_out;

    hipLaunchKernelGGL(k1_front, dim3(1), dim3(256), 0, stream,
                       audio, fb, adapt, ws_f);
    hipLaunchKernelGGL(k2_wmma_matvec, dim3(OUTN / 128), dim3(256), 0, stream,
                       W, bias, ws_f);
    hipLaunchKernelGGL(k3_spikes, dim3(OUTN / 4 / 256), dim3(256), 0, stream,
                       ws_f, out);
}